// GaussianModel_54975581389340
// MI455X (gfx1250) — compile-verified
//
#include <hip/hip_runtime.h>
#include <hip/hip_bf16.h>
#include <stdint.h>

#define GPB 4                  // gaussians per thread
#define TPB 256                // threads per block (8 wave32s)
#define CPB (GPB * TPB)        // gaussians per block chunk = 1024

typedef float v4f __attribute__((ext_vector_type(4)));
typedef int   v4i __attribute__((ext_vector_type(4)));

#if defined(__gfx1250__) && \
    __has_builtin(__builtin_amdgcn_global_load_async_to_lds_b128) && \
    __has_builtin(__builtin_amdgcn_s_wait_asynccnt)
#define USE_ASYNC 1
typedef __attribute__((address_space(1))) v4i* gptr_t;   // global int4*
typedef __attribute__((address_space(3))) v4i* lptr_t;   // LDS int4* (32-bit ptr)

// generic -> AS1 pointer: same 64-bit numeric value
__device__ __forceinline__ gptr_t to_global_v4i(const void* p) {
    return (gptr_t)(uintptr_t)p;
}
// generic LDS pointer -> AS3 pointer: hardware/addrspacecast truncates to addr[31:0]
__device__ __forceinline__ lptr_t to_lds_v4i(const void* p) {
    return (lptr_t)(uint32_t)(uintptr_t)p;
}
#else
#define USE_ASYNC 0
#endif

// Sigma = R diag(s) (R diag(s))^T for one gaussian. q = (w,x,y,z) Hamilton.
__device__ __forceinline__ void cov3d(float w, float x, float y, float z,
                                      float a, float b, float c, float* o) {
    float d = w * w + x * x + y * y + z * z;
    float inv = rsqrtf(d);
    inv = inv * (1.5f - 0.5f * d * inv * inv);   // one NR refinement
    w *= inv; x *= inv; y *= inv; z *= inv;
    float sx = __expf(a), sy = __expf(b), sz = __expf(c);
    float xx = x * x, yy = y * y, zz = z * z;
    float xy = x * y, xz = x * z, yz = y * z;
    float wx = w * x, wy = w * y, wz = w * z;
    float m00 = (1.f - 2.f * (yy + zz)) * sx, m01 = (2.f * (xy - wz)) * sy, m02 = (2.f * (xz + wy)) * sz;
    float m10 = (2.f * (xy + wz)) * sx, m11 = (1.f - 2.f * (xx + zz)) * sy, m12 = (2.f * (yz - wx)) * sz;
    float m20 = (2.f * (xz - wy)) * sx, m21 = (2.f * (yz + wx)) * sy, m22 = (1.f - 2.f * (xx + yy)) * sz;
    float s00 = m00 * m00 + m01 * m01 + m02 * m02;
    float s01 = m00 * m10 + m01 * m11 + m02 * m12;
    float s02 = m00 * m20 + m01 * m21 + m02 * m22;
    float s11 = m10 * m10 + m11 * m11 + m12 * m12;
    float s12 = m10 * m20 + m11 * m21 + m12 * m22;
    float s22 = m20 * m20 + m21 * m21 + m22 * m22;
    o[0] = s00; o[1] = s01; o[2] = s02;
    o[3] = s01; o[4] = s11; o[5] = s12;
    o[6] = s02; o[7] = s12; o[8] = s22;
}

__global__ __launch_bounds__(TPB) void gs_cov3d_kernel(const float* __restrict__ quats,
                                                       const float* __restrict__ scales,
                                                       float* __restrict__ out,
                                                       int n) {
#if USE_ASYNC
    __shared__ float qbuf[CPB * 4];   // 16 KB
    __shared__ float sbuf[CPB * 3];   // 12 KB
#endif
    const int t = threadIdx.x;
    const long long chunk0 = (long long)blockIdx.x * CPB;
    const long long base = chunk0 + (long long)t * GPB;
    const bool full = (chunk0 + CPB) <= (long long)n;   // uniform per block

    if (full) {
        float res[GPB * 9];
#if USE_ASYNC
        {
            const float* gq = quats + base * 4;          // 64 B, 16B-aligned
            const float* gs = scales + base * 3;         // 48 B, 16B-aligned
            float* lq = &qbuf[t * (GPB * 4)];
            float* ls = &sbuf[t * (GPB * 3)];
#pragma unroll
            for (int k = 0; k < 4; ++k)
                __builtin_amdgcn_global_load_async_to_lds_b128(
                    to_global_v4i(gq + 4 * k), to_lds_v4i(lq + 4 * k), 0, 0);
#pragma unroll
            for (int k = 0; k < 3; ++k)
                __builtin_amdgcn_global_load_async_to_lds_b128(
                    to_global_v4i(gs + 4 * k), to_lds_v4i(ls + 4 * k), 0, 0);
            __builtin_amdgcn_s_wait_asynccnt(0);
            __asm__ volatile("" ::: "memory");           // keep LDS reads after the wait
#pragma unroll
            for (int g = 0; g < GPB; ++g)
                cov3d(lq[4 * g + 0], lq[4 * g + 1], lq[4 * g + 2], lq[4 * g + 3],
                      ls[3 * g + 0], ls[3 * g + 1], ls[3 * g + 2], &res[9 * g]);
        }
#else
        {
            v4f qv[GPB];
            const v4f* q4 = (const v4f*)(quats + base * 4);
#pragma unroll
            for (int g = 0; g < GPB; ++g) qv[g] = q4[g];
            v4f sv[3];
            const v4f* s4 = (const v4f*)(scales + base * 3);
#pragma unroll
            for (int k = 0; k < 3; ++k) sv[k] = s4[k];
            const float* sf = (const float*)&sv[0];
#pragma unroll
            for (int g = 0; g < GPB; ++g)
                cov3d(qv[g].x, qv[g].y, qv[g].z, qv[g].w,
                      sf[3 * g + 0], sf[3 * g + 1], sf[3 * g + 2], &res[9 * g]);
        }
#endif
        // 36 contiguous floats per thread, 144B-aligned -> 9 x b128 NT stores
        float* ob = out + base * 9;
#pragma unroll
        for (int k = 0; k < 9; ++k) {
            v4f v = *(const v4f*)&res[4 * k];
            __builtin_nontemporal_store(v, (v4f*)(ob + 4 * k));
        }
    } else {
        // tail block: guarded scalar path
#pragma unroll
        for (int g = 0; g < GPB; ++g) {
            long long i = base + g;
            if (i < (long long)n) {
                float o[9];
                cov3d(quats[i * 4 + 0], quats[i * 4 + 1], quats[i * 4 + 2], quats[i * 4 + 3],
                      scales[i * 3 + 0], scales[i * 3 + 1], scales[i * 3 + 2], o);
                for (int k = 0; k < 9; ++k) out[i * 9 + k] = o[k];
            }
        }
    }
}

extern "C" void kernel_launch(void* const* d_in, const int* in_sizes, int n_in,
                              void* d_out, int out_size, void* d_ws, size_t ws_size,
                              hipStream_t stream) {
    const float* quats  = (const float*)d_in[0];   // (N,4) fp32
    const float* scales = (const float*)d_in[1];   // (N,3) fp32
    float* out = (float*)d_out;                    // (N,3,3) fp32
    int n = in_sizes[0] / 4;
    int blocks = (n + CPB - 1) / CPB;
    gs_cov3d_kernel<<<blocks, TPB, 0, stream>>>(quats, scales, out, n);
}